// GAT_42872363549079
// MI455X (gfx1250) — compile-verified
//
#include <hip/hip_runtime.h>
#include <hip/hip_bf16.h>

typedef __attribute__((ext_vector_type(2))) float v2f;
typedef __attribute__((ext_vector_type(8))) float v8f;

#define NNODE 50000
#define NEDGE 800000
#define ETOT  (NEDGE + NNODE)   // self-loops appended
#define NHEAD 4
#define CH    64                // per-head channels (both layers)
#define NC    (NHEAD * CH)      // 256 gemm output columns

// ---------- helpers ----------

// order-preserving float -> uint key (for atomic max on floats)
__device__ __forceinline__ unsigned fkey(float x) {
    unsigned u = __float_as_uint(x);
    return (u & 0x80000000u) ? ~u : (u | 0x80000000u);
}
__device__ __forceinline__ float funkey(unsigned k) {
    unsigned u = (k & 0x80000000u) ? (k & 0x7FFFFFFFu) : ~k;
    return __uint_as_float(u);
}

__device__ __forceinline__ void edge_nodes(const long long* __restrict__ ei,
                                           int e, int& s, int& d) {
    if (e < NEDGE) { s = (int)ei[e]; d = (int)ei[NEDGE + e]; }
    else           { s = e - NEDGE; d = s; }     // self loop
}

__device__ __forceinline__ float edge_e(const float* __restrict__ aS,
                                        const float* __restrict__ aD,
                                        int s, int d, int hd) {
    float e = aS[s * NHEAD + hd] + aD[d * NHEAD + hd];
    return e > 0.f ? e : 0.2f * e;               // leaky_relu(0.2)
}

// ---------- GEMM: C[M,256] = A[M,K] * W[256,K]^T via V_WMMA_F32_16X16X4_F32 ----------
// One wave per 16x16 output tile, 4 waves per block. M = 50000 = 3125*16 exact.
__global__ void __launch_bounds__(128)
gat_gemm_f32(const float* __restrict__ A, const float* __restrict__ W,
             float* __restrict__ C, int K)
{
    const int lane = threadIdx.x & 31;
    const int wave = threadIdx.x >> 5;
    const int tile = blockIdx.x * 4 + wave;      // 0 .. 49999
    const int mt = tile >> 4;                    // node-tile   0..3124
    const int nt = tile & 15;                    // column-tile 0..15
    const int row = lane & 15;
    const int kh  = lane >> 4;                   // 0: K=0,1  1: K=2,3

    const float* ap = A + (size_t)(mt * 16 + row) * K + 2 * kh;
    const float* bp = W + (size_t)(nt * 16 + row) * K + 2 * kh;

    v8f acc = {};
    for (int k0 = 0; k0 < K; k0 += 4) {
        v2f a = *reinterpret_cast<const v2f*>(ap + k0);
        v2f b = *reinterpret_cast<const v2f*>(bp + k0);
        acc = __builtin_amdgcn_wmma_f32_16x16x4_f32(
            /*neg_a=*/false, a, /*neg_b=*/false, b,
            /*c_mod=*/(short)0, acc, /*reuse_a=*/false, /*reuse_b=*/false);
    }

    // C/D layout: VGPR r -> (M = r + 8*kh, N = lane&15)
    float* cp = C + (size_t)(mt * 16 + 8 * kh) * NC + nt * 16 + row;
#pragma unroll
    for (int r = 0; r < 8; ++r)
        cp[(size_t)r * NC] = acc[r];
}

// ---------- per-(node,head) attention dots: one wave each ----------
__global__ void __launch_bounds__(128)
gat_alpha(const float* __restrict__ h, const float* __restrict__ aS,
          const float* __restrict__ aD, float* __restrict__ alphaS,
          float* __restrict__ alphaD)
{
    const int gw   = blockIdx.x * 4 + (threadIdx.x >> 5);  // global wave id
    const int lane = threadIdx.x & 31;
    const int node = gw >> 2;
    const int hd   = gw & 3;
    if (node >= NNODE) return;

    const float* hp = h  + (size_t)node * NC + hd * CH + 2 * lane;
    const float* sp = aS + hd * CH + 2 * lane;
    const float* dp = aD + hd * CH + 2 * lane;
    float ps = hp[0] * sp[0] + hp[1] * sp[1];
    float pd = hp[0] * dp[0] + hp[1] * dp[1];
#pragma unroll
    for (int off = 16; off; off >>= 1) {
        ps += __shfl_xor(ps, off, 32);
        pd += __shfl_xor(pd, off, 32);
    }
    if (lane == 0) {
        alphaS[node * NHEAD + hd] = ps;
        alphaD[node * NHEAD + hd] = pd;
    }
}

// ---------- edge pass 1: segment max (ordered-uint atomic max) ----------
__global__ void gat_emax(const long long* __restrict__ ei,
                         const float* __restrict__ aS, const float* __restrict__ aD,
                         unsigned* __restrict__ mkey)
{
    int idx = blockIdx.x * blockDim.x + threadIdx.x;
    if (idx >= ETOT * NHEAD) return;
    int e = idx >> 2, hd = idx & 3;
    int s, d; edge_nodes(ei, e, s, d);
    float v = edge_e(aS, aD, s, d, hd);
    atomicMax(&mkey[d * NHEAD + hd], fkey(v));
}

// ---------- edge pass 2: softmax denominator ----------
__global__ void gat_edenom(const long long* __restrict__ ei,
                           const float* __restrict__ aS, const float* __restrict__ aD,
                           const unsigned* __restrict__ mkey, float* __restrict__ denom)
{
    int idx = blockIdx.x * blockDim.x + threadIdx.x;
    if (idx >= ETOT * NHEAD) return;
    int e = idx >> 2, hd = idx & 3;
    int s, d; edge_nodes(ei, e, s, d);
    float v  = edge_e(aS, aD, s, d, hd);
    float ex = expf(v - funkey(mkey[d * NHEAD + hd]));
    atomicAdd(&denom[d * NHEAD + hd], ex);
}

// ---------- edge pass 3: weighted message scatter (one block per edge) ----------
__global__ void __launch_bounds__(256)
gat_scatter(const long long* __restrict__ ei,
            const float* __restrict__ aS, const float* __restrict__ aD,
            const unsigned* __restrict__ mkey, const float* __restrict__ denom,
            const float* __restrict__ h, float* __restrict__ acc)
{
    int e = blockIdx.x;
    int s, d; edge_nodes(ei, e, s, d);
    int hd = threadIdx.x >> 6;          // head 0..3
    int c  = threadIdx.x & 63;          // channel 0..63
    float v = edge_e(aS, aD, s, d, hd);
    float m = funkey(mkey[d * NHEAD + hd]);
    float alpha = expf(v - m) / (denom[d * NHEAD + hd] + 1e-16f);
    float msg = h[(size_t)s * NC + hd * CH + c] * alpha;
    atomicAdd(&acc[(size_t)d * NC + hd * CH + c], msg);
}

// ---------- finalize: head-mean + bias (+ ELU for layer 1) ----------
__global__ void gat_finalize(const float* __restrict__ acc, const float* __restrict__ b,
                             float* __restrict__ out, int apply_elu)
{
    int idx = blockIdx.x * blockDim.x + threadIdx.x;
    if (idx >= NNODE * CH) return;
    int n = idx >> 6, c = idx & 63;
    const float* a = acc + (size_t)n * NC + c;
    float v = (a[0] + a[CH] + a[2 * CH] + a[3 * CH]) * 0.25f + b[c];
    if (apply_elu) v = v > 0.f ? v : (expf(v) - 1.f);
    out[idx] = v;
}

// ---------- host ----------
extern "C" void kernel_launch(void* const* d_in, const int* in_sizes, int n_in,
                              void* d_out, int out_size, void* d_ws, size_t ws_size,
                              hipStream_t stream)
{
    const float*     x   = (const float*)d_in[0];
    const long long* ei  = (const long long*)d_in[1];     // [2, E] int64
    const float*     W1  = (const float*)d_in[2];
    const float*     as1 = (const float*)d_in[3];
    const float*     ad1 = (const float*)d_in[4];
    const float*     b1  = (const float*)d_in[5];
    const float*     W2  = (const float*)d_in[6];
    const float*     as2 = (const float*)d_in[7];
    const float*     ad2 = (const float*)d_in[8];
    const float*     b2  = (const float*)d_in[9];
    float*           out = (float*)d_out;

    // workspace layout
    float*    ws     = (float*)d_ws;
    float*    h      = ws;                                 // N*256
    float*    alphaS = h + (size_t)NNODE * NC;             // N*4
    float*    alphaD = alphaS + (size_t)NNODE * NHEAD;     // N*4
    unsigned* mkey   = (unsigned*)(alphaD + (size_t)NNODE * NHEAD); // N*4
    float*    denom  = (float*)mkey + (size_t)NNODE * NHEAD;        // N*4
    float*    acc    = denom + (size_t)NNODE * NHEAD;      // N*256
    float*    x2     = acc + (size_t)NNODE * NC;           // N*64 (layer1 output)

    const size_t clear_bytes = (size_t)NNODE * (NHEAD + NHEAD + NC) * sizeof(float);
    const int eth_blocks = (ETOT * NHEAD + 255) / 256;
    const int fin_blocks = (NNODE * CH + 255) / 256;

    for (int layer = 0; layer < 2; ++layer) {
        const float* xin = layer ? x2 : x;
        const int    K   = layer ? CH : 128;
        const float* W   = layer ? W2 : W1;
        const float* aS  = layer ? as2 : as1;
        const float* aD  = layer ? ad2 : ad1;
        const float* bv  = layer ? b2 : b1;
        float*       dst = layer ? out : x2;

        hipMemsetAsync(mkey, 0, clear_bytes, stream);      // mkey + denom + acc

        gat_gemm_f32<<<50000 / 4, 128, 0, stream>>>(xin, W, h, K);
        gat_alpha<<<NNODE * NHEAD / 4, 128, 0, stream>>>(h, aS, aD, alphaS, alphaD);
        gat_emax<<<eth_blocks, 256, 0, stream>>>(ei, alphaS, alphaD, mkey);
        gat_edenom<<<eth_blocks, 256, 0, stream>>>(ei, alphaS, alphaD, mkey, denom);
        gat_scatter<<<ETOT, 256, 0, stream>>>(ei, alphaS, alphaD, mkey, denom, h, acc);
        gat_finalize<<<fin_blocks, 256, 0, stream>>>(acc, bv, dst, layer == 0);
    }
}